// OverAll_74809740362204
// MI455X (gfx1250) — compile-verified
//
#include <hip/hip_runtime.h>

#define NODEN   200000
#define RELNUM  2000
#define TRIPLEN 800000
#define DIMN    128
#define OUTC    384          // concat of 3 layer outputs, row-major [NODE, 384]
#define EPSF    1e-12f

typedef __attribute__((ext_vector_type(2))) float v2f;
typedef __attribute__((ext_vector_type(8))) float v8f;

__device__ __forceinline__ void atomic_max_f32(float* addr, float val) {
  // int-punned float max: positives via signed max, negatives via unsigned min.
  if (val >= 0.0f) atomicMax((int*)addr, __float_as_int(val));
  else             atomicMin((unsigned int*)addr, __float_as_uint(val));
}

// out[:, 0:128) = tanh(features)
__global__ void k_tanh_in(const float* __restrict__ features, float* __restrict__ out) {
  int i = blockIdx.x * blockDim.x + threadIdx.x;
  if (i >= NODEN * DIMN) return;
  int n = i >> 7, k = i & 127;
  out[n * OUTC + k] = tanhf(features[i]);
}

// relnorm[c] = ||rel_emb[c]||2
__global__ void k_relnorm(const float* __restrict__ rel_emb, float* __restrict__ relnorm) {
  int c = blockIdx.x * blockDim.x + threadIdx.x;
  if (c >= RELNUM) return;
  const float* e = rel_emb + c * DIMN;
  float acc = 0.f;
#pragma unroll 4
  for (int k = 0; k < DIMN; ++k) acc += e[k] * e[k];
  relnorm[c] = sqrtf(acc);
}

// GEMV rel_dot[c] = rel_emb[c] . ak via V_WMMA_F32_16X16X4_F32.
// One wave = 16 relations (A: 16x4 per step), ak lives in B column 0.
__global__ void __launch_bounds__(32)
k_reldot_wmma(const float* __restrict__ rel_emb, const float* __restrict__ ak,
              float* __restrict__ rel_dot) {
  int lane = threadIdx.x;                 // 0..31, wave32
  int tile = blockIdx.x;                  // 16 relations per tile
  int row  = tile * 16 + (lane & 15);
  bool hi  = lane >= 16;
  // Lanes 0 and 16 carry B column N=0; all other lanes contribute zero.
  float msk = ((lane & 15) == 0) ? 1.0f : 0.0f;   // single v_cndmask, loop-invariant
  // A layout (32-bit 16x4): lanes 0-15: K=0 (v0), K=1 (v1); lanes 16-31: K=2,K=3.
  const float* abase = rel_emb + row * DIMN + (hi ? 2 : 0);
  // B layout (4x16): v0: lane j = B[0][j], lane 16+j = B[2][j]; v1: B[1][j]/B[3][j].
  // Lane 0 needs ak[k0],ak[k0+1]; lane 16 needs ak[k0+2],ak[k0+3].
  const float* bbase = ak + (hi ? 2 : 0);
  v8f acc = {};
#pragma unroll
  for (int k0 = 0; k0 < DIMN; k0 += 4) {
    v2f a; a.x = abase[k0]; a.y = abase[k0 + 1];
    float bx = bbase[k0];          // unconditional contiguous load (b64)
    float by = bbase[k0 + 1];
    v2f b; b.x = bx * msk; b.y = by * msk;   // arithmetic mask, no control flow
    acc = __builtin_amdgcn_wmma_f32_16x16x4_f32(false, a, false, b, (short)0, acc,
                                                false, false);
  }
  // D column N=0: lane 0 holds M=0..7 (vgpr 0..7), lane 16 holds M=8..15.
  if (lane == 0) {
#pragma unroll
    for (int r = 0; r < 8; ++r) rel_dot[tile * 16 + r] = acc[r];
  } else if (lane == 16) {
#pragma unroll
    for (int r = 0; r < 8; ++r) rel_dot[tile * 16 + 8 + r] = acc[r];
  }
}

__global__ void k_init_nodes(float* __restrict__ m, float* __restrict__ s) {
  int n = blockIdx.x * blockDim.x + threadIdx.x;
  if (n >= NODEN) return;
  m[n] = -INFINITY;
  s[n] = 0.0f;
}

__global__ void k_zero_cols(float* __restrict__ out, int colOff) {
  int i = blockIdx.x * blockDim.x + threadIdx.x;
  if (i >= NODEN * DIMN) return;
  int n = i >> 7, k = i & 127;
  out[n * OUTC + colOff + k] = 0.0f;
}

// att[t] = s_t * (rel_emb[c].ak);  m[dst] = segment max
__global__ void k_att_max(const int* __restrict__ dst, const int* __restrict__ cols,
                          const float* __restrict__ val, const float* __restrict__ relnorm,
                          const float* __restrict__ rel_dot, float* __restrict__ att,
                          float* __restrict__ m) {
  int t = blockIdx.x * blockDim.x + threadIdx.x;
  if (t >= TRIPLEN) return;
  int c   = cols[t];
  float v = val[t];
  float st = v / fmaxf(v * relnorm[c], EPSF);   // matches normalize(v*e) scale
  float a  = st * rel_dot[c];
  att[t] = a;
  atomic_max_f32(&m[dst[t]], a);
}

// att[t] = exp(att[t]-m[dst]);  s[dst] = segment sum
__global__ void k_exp_sum(const int* __restrict__ dst, float* __restrict__ att,
                          const float* __restrict__ m, float* __restrict__ s) {
  int t = blockIdx.x * blockDim.x + threadIdx.x;
  if (t >= TRIPLEN) return;
  int d = dst[t];
  float e = expf(att[t] - m[d]);
  att[t] = e;
  atomicAdd(&s[d], e);
}

// One wave32 per triple: gather neighbor row, Householder-reflect across r_t,
// weight by softmax, atomic scatter-add into next layer's 128-col slice.
__global__ void k_scatter(const int* __restrict__ dst, const int* __restrict__ src,
                          const int* __restrict__ cols, const float* __restrict__ val,
                          const float* __restrict__ relnorm, const float* __restrict__ att,
                          const float* __restrict__ s, const float* __restrict__ rel_emb,
                          float* __restrict__ out, int colIn, int colOut) {
  int lane = threadIdx.x & 31;
  int t = blockIdx.x * (blockDim.x >> 5) + (threadIdx.x >> 5);
  if (t >= TRIPLEN) return;
  int d   = dst[t];
  int c   = cols[t];
  float v = val[t];
  float st = v / fmaxf(v * relnorm[c], EPSF);
  float w  = att[t] / s[d];
  int k = lane * 4;
  const float4 nb = *(const float4*)(out + src[t] * OUTC + colIn + k);
  const float4 ec = *(const float4*)(rel_emb + c * DIMN + k);
  float p = nb.x * ec.x + nb.y * ec.y + nb.z * ec.z + nb.w * ec.w;
#pragma unroll
  for (int off = 16; off >= 1; off >>= 1) p += __shfl_xor(p, off, 32);
  float coef = 2.0f * st * st * p;   // neighs - 2*(neighs.r)*r, r = st*e_c
  float4 r;
  r.x = w * (nb.x - coef * ec.x);
  r.y = w * (nb.y - coef * ec.y);
  r.z = w * (nb.z - coef * ec.z);
  r.w = w * (nb.w - coef * ec.w);
  float* ob = out + d * OUTC + colOut + k;
  atomicAdd(ob + 0, r.x);
  atomicAdd(ob + 1, r.y);
  atomicAdd(ob + 2, r.z);
  atomicAdd(ob + 3, r.w);
}

__global__ void k_tanh_layer(float* __restrict__ out, int colOff) {
  int i = blockIdx.x * blockDim.x + threadIdx.x;
  if (i >= NODEN * DIMN) return;
  int n = i >> 7, k = i & 127;
  float* p = out + n * OUTC + colOff + k;
  *p = tanhf(*p);
}

extern "C" void kernel_launch(void* const* d_in, const int* in_sizes, int n_in,
                              void* d_out, int out_size, void* d_ws, size_t ws_size,
                              hipStream_t stream) {
  (void)in_sizes; (void)n_in; (void)out_size; (void)ws_size;
  const float* features   = (const float*)d_in[0];
  const float* rel_emb    = (const float*)d_in[1];
  const float* attn       = (const float*)d_in[2];   // [2,1,128]
  const float* sparse_val = (const float*)d_in[3];
  const int*   adj_dst    = (const int*)d_in[4];
  const int*   adj_src    = (const int*)d_in[5];
  const int*   sparse_cols= (const int*)d_in[7];     // sparse_rows (d_in[6]) == identity
  float* out = (float*)d_out;

  float* wsf     = (float*)d_ws;                 // ~4.8 MB total
  float* m       = wsf;                          // [NODE]
  float* s       = wsf + NODEN;                  // [NODE]
  float* att     = wsf + 2 * NODEN;              // [TRIPLE]
  float* relnorm = att + TRIPLEN;                // [REL]
  float* rel_dot = relnorm + RELNUM;             // [REL]

  const int BLK = 256;
  int gridNE = (NODEN * DIMN + BLK - 1) / BLK;   // 100000
  int gridT  = (TRIPLEN + BLK - 1) / BLK;        // 3125
  int gridN  = (NODEN + BLK - 1) / BLK;
  int gridW  = (TRIPLEN + (BLK / 32) - 1) / (BLK / 32); // 1 wave per triple

  k_tanh_in<<<gridNE, BLK, 0, stream>>>(features, out);
  k_relnorm<<<(RELNUM + BLK - 1) / BLK, BLK, 0, stream>>>(rel_emb, relnorm);

  for (int l = 0; l < 2; ++l) {
    const float* ak = attn + l * DIMN;           // HEADS == 1
    k_reldot_wmma<<<RELNUM / 16, 32, 0, stream>>>(rel_emb, ak, rel_dot);
    k_init_nodes<<<gridN, BLK, 0, stream>>>(m, s);
    k_zero_cols<<<gridNE, BLK, 0, stream>>>(out, (l + 1) * DIMN);
    k_att_max<<<gridT, BLK, 0, stream>>>(adj_dst, sparse_cols, sparse_val,
                                         relnorm, rel_dot, att, m);
    k_exp_sum<<<gridT, BLK, 0, stream>>>(adj_dst, att, m, s);
    k_scatter<<<gridW, BLK, 0, stream>>>(adj_dst, adj_src, sparse_cols, sparse_val,
                                         relnorm, att, s, rel_emb, out,
                                         l * DIMN, (l + 1) * DIMN);
    k_tanh_layer<<<gridNE, BLK, 0, stream>>>(out, (l + 1) * DIMN);
  }
}